// FixedGraphSAGE_56066503082351
// MI455X (gfx1250) — compile-verified
//
#include <hip/hip_runtime.h>
#include <hip/hip_bf16.h>

typedef float v2f __attribute__((ext_vector_type(2)));
typedef float v8f __attribute__((ext_vector_type(8)));

#define DIN 128

// ---------------------------------------------------------------------------
// GEMM: H[N x DOUT] = X[N x 128] @ W[128 x DOUT], fp32 via V_WMMA_F32_16X16X4_F32
// Block = 256 threads (8 waves). Each wave computes one 16-row tile across all
// DOUT columns. W staged in LDS (DIN*DOUT*4 bytes <= 64KB << 320KB/WGP).
// ---------------------------------------------------------------------------
template <int DOUT>
__global__ __launch_bounds__(256) void gcn_gemm_wmma(
    const float* __restrict__ X, const float* __restrict__ W,
    float* __restrict__ H, int nRows) {
  constexpr int NT = DOUT / 16;
  __shared__ float Ws[DIN * DOUT];
  const int t = threadIdx.x;
  for (int i = t * 4; i < DIN * DOUT; i += 256 * 4) {
    *(float4*)(&Ws[i]) = *(const float4*)(&W[i]);
  }
  __syncthreads();

  const int wave = t >> 5;
  const int lane = t & 31;
  const int m0 = (blockIdx.x * 8 + wave) * 16;
  if (m0 >= nRows) return;  // wave-uniform: EXEC stays all-ones for WMMA

  const int mr = lane & 15;   // A: row within tile; B: column within tile
  const int kh = lane >> 4;   // K half selector (0/1)

  v8f acc[NT] = {};
  const float* xrow = X + (size_t)(m0 + mr) * DIN;

  for (int k0 = 0; k0 < DIN; k0 += 4) {
    const int kb = k0 + 2 * kh;
    v2f a = *(const v2f*)(xrow + kb);  // 8B-aligned (kb even)
#pragma unroll
    for (int nt = 0; nt < NT; ++nt) {
      v2f b;
      b.x = Ws[kb * DOUT + nt * 16 + mr];
      b.y = Ws[(kb + 1) * DOUT + nt * 16 + mr];
      acc[nt] = __builtin_amdgcn_wmma_f32_16x16x4_f32(
          false, a, false, b, (short)0, acc[nt], false, false);
    }
  }

  // C/D layout: VGPR v, lanes 0-15 -> M=v, lanes 16-31 -> M=v+8, N=lane&15
  const int rbase = m0 + 8 * kh;
#pragma unroll
  for (int nt = 0; nt < NT; ++nt) {
#pragma unroll
    for (int v = 0; v < 8; ++v) {
      H[(size_t)(rbase + v) * DOUT + nt * 16 + mr] = acc[nt][v];
    }
  }
}

// ---------------------------------------------------------------------------
// Degree / normalization precompute
// ---------------------------------------------------------------------------
__global__ void deg_init_k(float* deg, int n) {
  int i = blockIdx.x * blockDim.x + threadIdx.x;
  if (i < n) deg[i] = 1.0f;  // self-loop weight 1
}

__global__ void deg_acc_k(const int* __restrict__ dst,
                          const float* __restrict__ ew, float* deg, int E) {
  int i = blockIdx.x * blockDim.x + threadIdx.x;
  if (i < E) atomicAdd(&deg[dst[i]], ew[i]);
}

__global__ void deg_inv_k(float* deg, int n) {
  int i = blockIdx.x * blockDim.x + threadIdx.x;
  if (i < n) deg[i] = 1.0f / sqrtf(deg[i]);  // deg >= 1 always
}

__global__ void edge_norm_k(const int* __restrict__ src,
                            const int* __restrict__ dst,
                            const float* __restrict__ ew,
                            const float* __restrict__ dis,
                            float* __restrict__ nrm, int E) {
  int i = blockIdx.x * blockDim.x + threadIdx.x;
  if (i < E) nrm[i] = dis[src[i]] * ew[i] * dis[dst[i]];
}

// ---------------------------------------------------------------------------
// Aggregation: out[v] = bias + dis[v]^2 * h[v] + sum_{e: src->v} norm[e]*h[src]
// ---------------------------------------------------------------------------
template <int D>
__global__ void agg_init_k(const float* __restrict__ H,
                           const float* __restrict__ dis,
                           const float* __restrict__ bias,
                           float* __restrict__ out, int n) {
  size_t idx = (size_t)blockIdx.x * blockDim.x + threadIdx.x;
  if (idx >= (size_t)n * D) return;
  int v = (int)(idx / D);
  int f = (int)(idx % D);
  float s = dis[v];
  out[idx] = H[idx] * s * s + bias[f];
}

// one wave per edge; D=128: each lane handles a float4 (coalesced gather)
__global__ __launch_bounds__(256) void scatter128_k(
    const int* __restrict__ src, const int* __restrict__ dst,
    const float* __restrict__ nrm, const float* __restrict__ H,
    float* __restrict__ out, int E) {
  int wid = (int)((blockIdx.x * (size_t)blockDim.x + threadIdx.x) >> 5);
  int lane = threadIdx.x & 31;
  if (wid >= E) return;
  int s = src[wid], d = dst[wid];
  float w = nrm[wid];
  float4 h4 = *(const float4*)(H + (size_t)s * 128 + lane * 4);
  float* o = out + (size_t)d * 128 + lane * 4;
  atomicAdd(o + 0, h4.x * w);
  atomicAdd(o + 1, h4.y * w);
  atomicAdd(o + 2, h4.z * w);
  atomicAdd(o + 3, h4.w * w);
}

__global__ __launch_bounds__(256) void scatter64_k(
    const int* __restrict__ src, const int* __restrict__ dst,
    const float* __restrict__ nrm, const float* __restrict__ H,
    float* __restrict__ out, int E) {
  int wid = (int)((blockIdx.x * (size_t)blockDim.x + threadIdx.x) >> 5);
  int lane = threadIdx.x & 31;
  if (wid >= E) return;
  int s = src[wid], d = dst[wid];
  float w = nrm[wid];
  float2 h2 = *(const float2*)(H + (size_t)s * 64 + lane * 2);
  float* o = out + (size_t)d * 64 + lane * 2;
  atomicAdd(o + 0, h2.x * w);
  atomicAdd(o + 1, h2.y * w);
}

// ---------------------------------------------------------------------------
// BatchNorm (population stats) + leaky ReLU(0.1)
// ---------------------------------------------------------------------------
__global__ void zero_stats_k(float* stats) { stats[threadIdx.x] = 0.0f; }

__global__ __launch_bounds__(128) void bn_stats_k(const float* __restrict__ X,
                                                  float* __restrict__ stats,
                                                  int n) {
  int f = threadIdx.x;  // 0..127
  int r0 = blockIdx.x * 256;
  float s = 0.f, s2 = 0.f;
  int rend = min(r0 + 256, n);
  for (int r = r0; r < rend; ++r) {
    float v = X[(size_t)r * 128 + f];
    s += v;
    s2 += v * v;
  }
  atomicAdd(&stats[f], s);
  atomicAdd(&stats[128 + f], s2);
}

__global__ void bn_apply_k(float* __restrict__ X,
                           const float* __restrict__ stats,
                           const float* __restrict__ g,
                           const float* __restrict__ be, int n) {
  size_t idx = (size_t)blockIdx.x * blockDim.x + threadIdx.x;
  if (idx >= (size_t)n * 128) return;
  int f = (int)(idx & 127);
  float invn = 1.0f / (float)n;
  float mu = stats[f] * invn;
  float var = stats[128 + f] * invn - mu * mu;
  float y = (X[idx] - mu) * (1.0f / sqrtf(var + 1e-5f)) * g[f] + be[f];
  X[idx] = y >= 0.f ? y : 0.1f * y;
}

// ---------------------------------------------------------------------------
// L2 row-normalize (D=64), one wave per row, wave32 xor-shuffle reduction
// ---------------------------------------------------------------------------
__global__ __launch_bounds__(256) void l2norm_k(float* __restrict__ out, int n) {
  int wid = (int)((blockIdx.x * (size_t)blockDim.x + threadIdx.x) >> 5);
  int lane = threadIdx.x & 31;
  if (wid >= n) return;
  float* row = out + (size_t)wid * 64;
  float2 v = *(float2*)(row + lane * 2);
  float s = v.x * v.x + v.y * v.y;
#pragma unroll
  for (int off = 16; off > 0; off >>= 1) s += __shfl_xor(s, off, 32);
  float inv = 1.0f / fmaxf(sqrtf(s), 1e-12f);
  v.x *= inv;
  v.y *= inv;
  *(float2*)(row + lane * 2) = v;
}

// ---------------------------------------------------------------------------
extern "C" void kernel_launch(void* const* d_in, const int* in_sizes, int n_in,
                              void* d_out, int out_size, void* d_ws,
                              size_t ws_size, hipStream_t stream) {
  const float* x = (const float*)d_in[0];
  const int* ei = (const int*)d_in[1];
  const float* ew = (const float*)d_in[2];
  const float* W1 = (const float*)d_in[3];
  const float* b1 = (const float*)d_in[4];
  const float* g1 = (const float*)d_in[5];
  const float* be1 = (const float*)d_in[6];
  const float* W2 = (const float*)d_in[7];
  const float* b2 = (const float*)d_in[8];
  const float* g2 = (const float*)d_in[9];
  const float* be2 = (const float*)d_in[10];
  const float* W3 = (const float*)d_in[11];
  const float* b3 = (const float*)d_in[12];

  const int N = in_sizes[0] / 128;  // 100000
  const int E = in_sizes[2];        // 1600000
  const int* src = ei;
  const int* dst = ei + E;

  float* ws = (float*)d_ws;
  float* dis = ws;                             // N
  float* nrm = dis + N;                        // E
  float* bufA = nrm + E;                       // N*128
  float* bufB = bufA + (size_t)N * 128;        // N*128
  float* stats = bufB + (size_t)N * 128;       // 256
  float* out = (float*)d_out;                  // N*64

  const int TB = 256;
  const int nBlkN = (N + TB - 1) / TB;
  const int nBlkE = (E + TB - 1) / TB;
  const int nBlkNF128 = (int)(((size_t)N * 128 + TB - 1) / TB);
  const int nBlkNF64 = (int)(((size_t)N * 64 + TB - 1) / TB);
  const int gemmBlk = ((N + 15) / 16 + 7) / 8;
  const int edgeWaveBlk = (E + 7) / 8;   // one wave (32 lanes) per edge
  const int rowWaveBlk = (N + 7) / 8;    // one wave per row
  const int bnBlk = (N + 255) / 256;

  // --- degree / symmetric normalization ---
  deg_init_k<<<nBlkN, TB, 0, stream>>>(dis, N);
  deg_acc_k<<<nBlkE, TB, 0, stream>>>(dst, ew, dis, E);
  deg_inv_k<<<nBlkN, TB, 0, stream>>>(dis, N);
  edge_norm_k<<<nBlkE, TB, 0, stream>>>(src, dst, ew, dis, nrm, E);

  // --- layer 1: GCNConv(128->128) + BN + leakyReLU ---
  gcn_gemm_wmma<128><<<gemmBlk, 256, 0, stream>>>(x, W1, bufA, N);
  agg_init_k<128><<<nBlkNF128, TB, 0, stream>>>(bufA, dis, b1, bufB, N);
  scatter128_k<<<edgeWaveBlk, 256, 0, stream>>>(src, dst, nrm, bufA, bufB, E);
  zero_stats_k<<<1, 256, 0, stream>>>(stats);
  bn_stats_k<<<bnBlk, 128, 0, stream>>>(bufB, stats, N);
  bn_apply_k<<<nBlkNF128, TB, 0, stream>>>(bufB, stats, g1, be1, N);

  // --- layer 2: GCNConv(128->128) + BN + leakyReLU ---
  gcn_gemm_wmma<128><<<gemmBlk, 256, 0, stream>>>(bufB, W2, bufA, N);
  agg_init_k<128><<<nBlkNF128, TB, 0, stream>>>(bufA, dis, b2, bufB, N);
  scatter128_k<<<edgeWaveBlk, 256, 0, stream>>>(src, dst, nrm, bufA, bufB, E);
  zero_stats_k<<<1, 256, 0, stream>>>(stats);
  bn_stats_k<<<bnBlk, 128, 0, stream>>>(bufB, stats, N);
  bn_apply_k<<<nBlkNF128, TB, 0, stream>>>(bufB, stats, g2, be2, N);

  // --- layer 3: GCNConv(128->64) + row L2 normalize ---
  gcn_gemm_wmma<64><<<gemmBlk, 256, 0, stream>>>(bufB, W3, bufA, N);
  agg_init_k<64><<<nBlkNF64, TB, 0, stream>>>(bufA, dis, b3, out, N);
  scatter64_k<<<edgeWaveBlk, 256, 0, stream>>>(src, dst, nrm, bufA, out, E);
  l2norm_k<<<rowWaveBlk, 256, 0, stream>>>(out, N);
}